// SE3PointKernel_71657234366701
// MI455X (gfx1250) — compile-verified
//
#include <hip/hip_runtime.h>
#include <hip/hip_bf16.h>
#include <math.h>

typedef __attribute__((ext_vector_type(2))) float v2f;
typedef __attribute__((ext_vector_type(8))) float v8f;

#define NPTS        65536      // B*N*M
#define PTS_PER_WG  64
#define NRAD        304
#define HID         64
#define QTOT        1225
#define H_STRIDE    68         // 64 + pad (bank-conflict-free fragment loads)
#define W_STRIDE    305        // 304 + pad (stride mod 64 coprime -> conflict-free)
#define Y_STRIDE    25

static __device__ __forceinline__ int imax(int a,int b){return a>b?a:b;}
static __device__ __forceinline__ int imin(int a,int b){return a<b?a:b;}

// ---------------------------------------------------------------------------
// Q-basis init: real-basis Clebsch-Gordan intertwiner, unit Frobenius norm.
// This equals (up to global sign) the SVD null-vector Q of the reference.
// Recomputed every launch (deterministic, idempotent) into d_ws.
// ---------------------------------------------------------------------------
__device__ double dfact(int n){ double f=1.0; for(int i=2;i<=n;++i) f*=(double)i; return f; }

__device__ double cgcoef(int l1,int m1,int l2,int m2,int L,int M){
  if (m1+m2 != M) return 0.0;
  if (L < abs(l1-l2) || L > l1+l2) return 0.0;
  if (abs(m1)>l1 || abs(m2)>l2 || abs(M)>L) return 0.0;
  double pref = sqrt((2.0*L+1.0)*dfact(L+l1-l2)*dfact(L-l1+l2)*dfact(l1+l2-L)/dfact(l1+l2+L+1));
  pref *= sqrt(dfact(L+M)*dfact(L-M)*dfact(l1-m1)*dfact(l1+m1)*dfact(l2-m2)*dfact(l2+m2));
  int kmin = imax(0, imax(l2-L-m1, l1+m2-L));
  int kmax = imin(l1+l2-L, imin(l1-m1, l2+m2));
  double s = 0.0;
  for (int k=kmin;k<=kmax;++k){
    double d = dfact(k)*dfact(l1+l2-L-k)*dfact(l1-m1-k)*dfact(l2+m2-k)
             * dfact(L-l2+m1+k)*dfact(L-l1-m2+k);
    s += ((k&1)? -1.0:1.0)/d;
  }
  return pref*s;
}

// U[mp+l, m+l] for real<-complex SH change of basis (reference _U)
__device__ void Uelem(int l,int mp,int m,double&re,double&im){
  const double s2 = 0.70710678118654752;
  re = 0.0; im = 0.0;
  if (mp > 0){
    if (m == -mp) re = s2;
    else if (m == mp) re = (mp & 1) ? -s2 : s2;
  } else if (mp == 0){
    if (m == 0) re = 1.0;
  } else {
    int am = -mp;
    if (m == mp) im = s2;
    else if (m == am) im = (am & 1) ? s2 : -s2;
  }
}

__global__ void se3_init_q(float* __restrict__ Qout){
  int t = threadIdx.x;
  if (t >= 19) return;
  const int TB[19][4] = {
    {0,0,0,   0},{0,1,1,   1},{0,2,2,  10},{1,0,1,  35},
    {1,1,0,  44},{1,1,1,  53},{1,1,2,  80},
    {1,2,1, 125},{1,2,2, 170},{1,2,3, 245},
    {2,0,2, 350},
    {2,1,1, 375},{2,1,2, 420},{2,1,3, 495},
    {2,2,0, 600},{2,2,1, 625},{2,2,2, 700},{2,2,3, 825},{2,2,4,1000}};
  int lo = TB[t][0], li = TB[t][1], J = TB[t][2], off = TB[t][3];
  int ni = 2*lo+1, nj = 2*li+1, nr = 2*J+1;
  int cnt = ni*nj*nr;
  double Tre[225], Tim[225];
  int idx = 0;
  for (int i=0;i<ni;++i)
  for (int j=0;j<nj;++j)
  for (int r=0;r<nr;++r){
    double are=0.0, aim=0.0;
    for (int mo=-lo;mo<=lo;++mo){
      double u1r,u1i; Uelem(lo, i-lo, mo, u1r,u1i);
      if (u1r==0.0 && u1i==0.0) continue;
      for (int mi=-li;mi<=li;++mi){
        double u2r,u2i; Uelem(li, j-li, mi, u2r,u2i);
        if (u2r==0.0 && u2i==0.0) continue;
        int mJ = mo+mi;
        if (mJ < -J || mJ > J) continue;
        double u3r,u3i; Uelem(J, r-J, mJ, u3r,u3i);
        u3i = -u3i;                                  // conjugate
        double c = cgcoef(lo,mo,li,mi,J,mJ);
        if (c==0.0) continue;
        double pr = u1r*u2r - u1i*u2i;
        double pi = u1r*u2i + u1i*u2r;
        are += (pr*u3r - pi*u3i)*c;
        aim += (pr*u3i + pi*u3r)*c;
      }
    }
    Tre[idx]=are; Tim[idx]=aim; ++idx;
  }
  // remove global phase, take real part, normalize to unit Frobenius norm
  double best=0.0, br=1.0, bi=0.0;
  for (int k=0;k<cnt;++k){
    double m2 = Tre[k]*Tre[k] + Tim[k]*Tim[k];
    if (m2 > best){ best=m2; br=Tre[k]; bi=Tim[k]; }
  }
  double mag = sqrt(best);
  double cr = br/mag, ci = -bi/mag;                  // e^{-i theta}
  double nrm = 0.0;
  for (int k=0;k<cnt;++k){ double re = Tre[k]*cr - Tim[k]*ci; nrm += re*re; }
  nrm = 1.0/sqrt(nrm);
  for (int k=0;k<cnt;++k){
    double re = Tre[k]*cr - Tim[k]*ci;
    Qout[off+k] = (float)(re*nrm);
  }
}

// ---------------------------------------------------------------------------
// Fused main kernel: 1 workgroup (256 thr = 8 wave32) = 64 consecutive points.
// Phase 1 GEMM uses V_WMMA_F32_16X16X4_F32 (exact f32 path, wave32 fragments).
// ---------------------------------------------------------------------------
__global__ __launch_bounds__(256) void se3_main(
    const float* __restrict__ diff, const float* __restrict__ W1,
    const float* __restrict__ b1,  const float* __restrict__ W2,
    const float* __restrict__ b2,  const float* __restrict__ Qg,
    float* __restrict__ out)
{
  extern __shared__ float smem[];
  float* sH = smem;                              // 64 x 68
  float* sW = smem + PTS_PER_WG*H_STRIDE;        // 64 x 305
  float* sY = sW   + PTS_PER_WG*W_STRIDE;        // 64 x 25
  float* sQ = sY   + PTS_PER_WG*Y_STRIDE;        // 1225

  const int tid = threadIdx.x;

  // prefetch W2 (L2-resident across all 1024 workgroups)
  __builtin_prefetch(W2 + tid*76, 0, 1);

  for (int q = tid; q < QTOT; q += 256) sQ[q] = Qg[q];

  // --- Phase 0: radial MLP layer-1 (H) + spherical harmonics (Y) -----------
  {
    const int p = tid & 63;
    const size_t pg = (size_t)blockIdx.x * PTS_PER_WG + p;
    const float x = diff[pg*3+0], y = diff[pg*3+1], z = diff[pg*3+2];
    const float r = sqrtf(x*x + y*y + z*z);
    const int j0 = (tid >> 6) * 16;                // 4 threads/point, 16 hidden each
#pragma unroll
    for (int jj=0; jj<16; ++jj){
      int j = j0 + jj;
      float h = fmaf(r, W1[j], b1[j]);
      sH[p*H_STRIDE + j] = h > 0.f ? h : 0.f;
    }
    if (tid < 64){
      float reps = r + 1e-12f;
      float ct = z / reps;
      float st2 = fmaxf(1.f - ct*ct, 0.f);
      float st = sqrtf(st2);
      float rho = sqrtf(x*x + y*y);
      float c1 = 1.f, s1 = 0.f;
      if (rho > 0.f){ c1 = x/rho; s1 = y/rho; }
      float c2 = c1*c1 - s1*s1, s2 = 2.f*c1*s1;
      float c3 = c2*c1 - s2*s1, s3 = s2*c1 + c2*s1;
      float c4 = c3*c1 - s3*s1, s4 = s3*c1 + c3*s1;
      float P00 = 1.f;
      float P10 = ct,        P11 = st;
      float P20 = 0.5f*(3.f*ct*P10 - P00);
      float P21 = 3.f*ct*st, P22 = 3.f*st2;
      float P30 = (5.f*ct*P20 - 2.f*P10)*(1.f/3.f);
      float P31 = (5.f*ct*P21 - 3.f*P11)*0.5f;
      float P32 = 15.f*ct*st2, P33 = 15.f*st*st2;
      float P40 = (7.f*ct*P30 - 3.f*P20)*0.25f;
      float P41 = (7.f*ct*P31 - 4.f*P21)*(1.f/3.f);
      float P42 = (7.f*ct*P32 - 5.f*P22)*0.5f;
      float P43 = 7.f*ct*P33, P44 = 7.f*st*P33;
      const float SQ2 = 1.41421356237f;
      float* Yp = &sY[tid*Y_STRIDE];
      Yp[0]  = 0.28209479177f*P00;
      Yp[1]  = SQ2*0.34549414947f*P11*s1;
      Yp[2]  = 0.48860251190f*P10;
      Yp[3]  = SQ2*0.34549414947f*P11*c1;
      Yp[4]  = SQ2*0.12875806734f*P22*s2;
      Yp[5]  = SQ2*0.25751613468f*P21*s1;
      Yp[6]  = 0.63078313051f*P20;
      Yp[7]  = SQ2*0.25751613468f*P21*c1;
      Yp[8]  = SQ2*0.12875806734f*P22*c2;
      Yp[9]  = SQ2*0.02781492158f*P33*s3;
      Yp[10] = SQ2*0.06813236510f*P32*s2;
      Yp[11] = SQ2*0.21545345608f*P31*s1;
      Yp[12] = 0.74635266518f*P30;
      Yp[13] = SQ2*0.21545345608f*P31*c1;
      Yp[14] = SQ2*0.06813236510f*P32*c2;
      Yp[15] = SQ2*0.02781492158f*P33*c3;
      Yp[16] = SQ2*0.00421459707f*P44*s4;
      Yp[17] = SQ2*0.01192068068f*P43*s3;
      Yp[18] = SQ2*0.04460301906f*P42*s2;
      Yp[19] = SQ2*0.18923493915f*P41*s1;
      Yp[20] = 0.84628437532f*P40;
      Yp[21] = SQ2*0.18923493915f*P41*c1;
      Yp[22] = SQ2*0.04460301906f*P42*c2;
      Yp[23] = SQ2*0.01192068068f*P43*c3;
      Yp[24] = SQ2*0.00421459707f*P44*c4;
    }
  }
  __syncthreads();

  // --- Phase 1: weights[64x304] = H[64x64] @ W2[64x304] + b2 via WMMA ------
  {
    const int wave = tid >> 5, lane = tid & 31;
    const int l15 = lane & 15;
    const int khalf = (lane >> 4) << 1;            // lanes 16-31 hold K+2/K+3
    for (int tile = wave; tile < 76; tile += 8){   // 4 M-tiles x 19 N-tiles
      const int mt = tile / 19, nt = tile % 19;
      const int rowA = mt*16 + l15;                // point row
      const int colB = nt*16 + l15;                // radial column
      const float* hrow = &sH[rowA*H_STRIDE];
      v8f acc = {0.f,0.f,0.f,0.f,0.f,0.f,0.f,0.f};
#pragma unroll
      for (int k0 = 0; k0 < HID; k0 += 4){
        v2f a = { hrow[k0+khalf], hrow[k0+khalf+1] };
        v2f b = { W2[(size_t)(k0+khalf)*NRAD + colB],
                  W2[(size_t)(k0+khalf+1)*NRAD + colB] };
        acc = __builtin_amdgcn_wmma_f32_16x16x4_f32(
                  false, a, false, b, (short)0, acc, false, false);
      }
      const int crow = mt*16 + ((lane >> 4) << 3); // C: vgpr v -> M=v (+8 hi half)
      const float bias = b2[colB];
#pragma unroll
      for (int v = 0; v < 8; ++v)
        sW[(crow+v)*W_STRIDE + colB] = acc[v] + bias;
    }
  }
  __syncthreads();

  // --- Phase 2: per-point equivariant kernel assembly + coalesced store ----
  {
    const int p = tid & 63;                        // lane-consecutive points
    const int g = tid >> 6;                        // 4 (i,j)-slices
    const size_t pointg = (size_t)blockIdx.x * PTS_PER_WG + p;  // == n*256+m
    const float* Yp = &sY[p*Y_STRIDE];
    const float* Wp = &sW[p*W_STRIDE];

    constexpr int pair_lo[9]    = {0,0,0,1,1,1,2,2,2};
    constexpr int pair_li[9]    = {0,1,2,0,1,2,0,1,2};
    constexpr int pair_Jmin[9]  = {0,1,2,1,0,1,2,1,0};
    constexpr int pair_nJ[9]    = {1,1,1,1,3,3,1,3,5};
    constexpr int pair_wb[9]    = {0,16,32,48,64,112,160,176,224};
    constexpr int pair_qb[9]    = {0,1,10,35,44,125,350,375,600};
    constexpr int base_l[3]     = {0,4,16};        // row/col start per l

#pragma unroll
    for (int pi = 0; pi < 9; ++pi){
      const int lo = pair_lo[pi], li = pair_li[pi];
      const int ni = 2*lo+1, njd = 2*li+1, nij = ni*njd;
      const int nJ = pair_nJ[pi], Jmin = pair_Jmin[pi];
      for (int ij = g; ij < nij; ij += 4){
        const int i = ij / njd, j = ij % njd;
        float K[5];
        int qoff = pair_qb[pi];
#pragma unroll
        for (int Jx = 0; Jx < nJ; ++Jx){
          const int J = Jmin + Jx, dim = 2*J+1;
          const float* Qb = &sQ[qoff + ij*dim];    // lane-uniform -> broadcast
          const float* Yb = &Yp[J*J];              // offset of Y_J == J^2
          float acc = 0.f;
#pragma unroll
          for (int rr = 0; rr < dim; ++rr) acc = fmaf(Qb[rr], Yb[rr], acc);
          K[Jx] = acc;
          qoff += nij*dim;
        }
        const int row0 = base_l[lo] + i;
        const int col0 = base_l[li] + j;
#pragma unroll
        for (int u = 0; u < 4; ++u)
#pragma unroll
          for (int v = 0; v < 4; ++v){
            const float* wv = &Wp[pair_wb[pi] + (u*4+v)*nJ];
            float val = 0.f;
#pragma unroll
            for (int Jx = 0; Jx < nJ; ++Jx) val = fmaf(wv[Jx], K[Jx], val);
            const int row = row0 + u*ni, col = col0 + v*njd;
            out[(size_t)(row*36 + col)*NPTS + pointg] = val;
          }
      }
    }
  }
}

// ---------------------------------------------------------------------------
extern "C" void kernel_launch(void* const* d_in, const int* in_sizes, int n_in,
                              void* d_out, int out_size, void* d_ws, size_t ws_size,
                              hipStream_t stream) {
  (void)in_sizes; (void)n_in; (void)out_size; (void)ws_size;
  const float* diff = (const float*)d_in[0];
  const float* W1   = (const float*)d_in[1];
  const float* b1   = (const float*)d_in[2];
  const float* W2   = (const float*)d_in[3];
  const float* b2   = (const float*)d_in[4];
  float* out = (float*)d_out;
  float* Q   = (float*)d_ws;                      // 1225 floats

  se3_init_q<<<1, 32, 0, stream>>>(Q);

  const int lds_floats = PTS_PER_WG*H_STRIDE + PTS_PER_WG*W_STRIDE
                       + PTS_PER_WG*Y_STRIDE + QTOT;   // 26697
  se3_main<<<NPTS/PTS_PER_WG, 256, lds_floats*(int)sizeof(float), stream>>>(
      diff, W1, b1, W2, b2, Q, out);
}